// MultiHeadAttention_76158360092785
// MI455X (gfx1250) — compile-verified
//
#include <hip/hip_runtime.h>

typedef float v2f __attribute__((ext_vector_type(2)));
typedef float v4f __attribute__((ext_vector_type(4)));
typedef float v8f __attribute__((ext_vector_type(8)));

static __device__ __forceinline__ v8f wmma4(v2f a, v2f b, v8f c) {
  // V_WMMA_F32_16X16X4_F32 : D = A(16x4,f32) * B(4x16,f32) + C(16x16,f32)
  return __builtin_amdgcn_wmma_f32_16x16x4_f32(false, a, false, b, (short)0, c,
                                               false, false);
}
static __device__ __forceinline__ v2f lo2(v4f v) {
  v2f r; r.x = v.x; r.y = v.y; return r;
}
static __device__ __forceinline__ v2f hi2(v4f v) {
  v2f r; r.x = v.z; r.y = v.w; return r;
}

// Wave-internal LDS RAW fence (cross-lane exchange through LDS within a wave).
#define WAVE_LDS_FENCE() asm volatile("s_wait_dscnt 0x0" ::: "memory")

// CDNA5 async memory->LDS copy (16B per lane), tracked by ASYNCcnt.
static __device__ __forceinline__ void async_cp16(float* lds_dst,
                                                  const float* gsrc) {
  unsigned off = (unsigned)(uintptr_t)lds_dst;  // low 32 bits = LDS offset
  asm volatile("global_load_async_to_lds_b128 %0, %1, off"
               :
               : "v"(off), "v"(gsrc)
               : "memory");
}
#define ASYNC_WAIT_ALL()  asm volatile("s_wait_asynccnt 0x0" ::: "memory")
#define ASYNC_WAIT_PREV() asm volatile("s_wait_asynccnt 0x4" ::: "memory")

static __device__ __forceinline__ v8f vzero8() {
  v8f z = {0.f, 0.f, 0.f, 0.f, 0.f, 0.f, 0.f, 0.f};
  return z;
}

// ---------------------------------------------------------------------------
// Stage 1: Out[M][N] = A[M][K] @ W[N][K]^T
// 64x16 strip per wave (4 m-tiles, B stationary).  Each b128 load provides
// TWO wmma steps (k-permuted lo/hi pairing); a sched_barrier separates the
// load phase from the WMMA phase so loads stay in flight.
// ---------------------------------------------------------------------------
__global__ __launch_bounds__(256) void gemm_xt(const float* __restrict__ A,
                                               const float* __restrict__ W,
                                               float* __restrict__ Out,
                                               int M, int N, int K) {
  const int lane = threadIdx.x & 31;
  const int wid  = threadIdx.x >> 5;
  const int ln   = lane & 15;  // n (B) / m (A) within tile
  const int hl   = lane >> 4;  // lane half

  const int ntiles = N >> 4;
  const int gw = blockIdx.x * 8 + wid;
  const int mt = gw / ntiles;  // 64-row strip index
  const int nt = gw % ntiles;

  const float* ar0 = A + (size_t)(mt * 64 + 0 * 16 + ln) * K;
  const float* ar1 = A + (size_t)(mt * 64 + 1 * 16 + ln) * K;
  const float* ar2 = A + (size_t)(mt * 64 + 2 * 16 + ln) * K;
  const float* ar3 = A + (size_t)(mt * 64 + 3 * 16 + ln) * K;
  const float* brow = W + (size_t)(nt * 16 + ln) * K;

  v8f acc[4] = {vzero8(), vzero8(), vzero8(), vzero8()};
  const int kblocks = K >> 3;  // 8 k-columns per block (one b128 per row)
#pragma unroll 2
  for (int kb = 0; kb < kblocks; kb += 2) {
    v4f bb[2];
    v4f aa[2][4];
#pragma unroll
    for (int s = 0; s < 2; ++s) {
      const int ko = (kb + s) * 8 + 4 * hl;
      bb[s]    = *(const v4f*)(brow + ko);
      aa[s][0] = *(const v4f*)(ar0 + ko);
      aa[s][1] = *(const v4f*)(ar1 + ko);
      aa[s][2] = *(const v4f*)(ar2 + ko);
      aa[s][3] = *(const v4f*)(ar3 + ko);
    }
    __builtin_amdgcn_sched_barrier(0);  // keep all 10 loads ahead of WMMAs
#pragma unroll
    for (int s = 0; s < 2; ++s) {
#pragma unroll
      for (int i = 0; i < 4; ++i)
        acc[i] = wmma4(lo2(aa[s][i]), lo2(bb[s]), acc[i]);
#pragma unroll
      for (int i = 0; i < 4; ++i)
        acc[i] = wmma4(hi2(aa[s][i]), hi2(bb[s]), acc[i]);
    }
  }

#pragma unroll
  for (int i = 0; i < 4; ++i) {
    float* orow = Out + (size_t)(mt * 64 + i * 16 + hl * 8) * N + nt * 16 + ln;
#pragma unroll
    for (int j = 0; j < 8; ++j) orow[(size_t)j * N] = acc[i][j];
  }
}

// ---------------------------------------------------------------------------
// Stage 2: flash attention per (b,h).  Q,K,V stored as [B*N][H*64] (col=h*64+d)
// K/V staged to LDS with double-buffered CDNA5 async-to-LDS copies (32-key
// chunks): chunk i+1's fetch overlaps chunk i's WMMA + softmax.
// Writes cat[B*N][512] with head-major columns (matches torch.cat(dim=2)).
// ---------------------------------------------------------------------------
__global__ __launch_bounds__(256) void flash_attn(const float* __restrict__ Q,
                                                  const float* __restrict__ Km,
                                                  const float* __restrict__ Vm,
                                                  float* __restrict__ Cat) {
  constexpr int N = 2048, D = 64, F = 512;
  constexpr int CH   = 32;  // keys per chunk
  constexpr int KSTR = 68;  // LDS row stride for K/V (16B aligned, low conflict)
  constexpr int PSTR = 20;  // LDS row stride for score tile (16B aligned)

  __shared__ float Kld[2][CH * KSTR];
  __shared__ float Vld[2][CH * KSTR];
  __shared__ float Pld[8][16 * PSTR];
  __shared__ float Stat[8][16];

  const int lane = threadIdx.x & 31;
  const int wid  = threadIdx.x >> 5;
  const int ln   = lane & 15;
  const int hl   = lane >> 4;

  const int bh   = blockIdx.x >> 4;  // 0..63
  const int qblk = blockIdx.x & 15;
  const int b    = bh >> 3;
  const int h    = bh & 7;
  const int qbase = qblk * 128 + wid * 16;

  // Q strip (16 rows x 64) as paired A-fragments, pre-scaled by 1/sqrt(64).
  const float* qrow_p = Q + (size_t)(b * N + qbase + ln) * F + h * D + 4 * hl;
  v4f qa4[8];
#pragma unroll
  for (int kb = 0; kb < 8; ++kb) {
    v4f t = *(const v4f*)(qrow_p + kb * 8);
    qa4[kb] = t * 0.125f;
  }

  v8f oacc[4];
#pragma unroll
  for (int vb = 0; vb < 4; ++vb) oacc[vb] = vzero8();
  float mi = -3.0e38f, li = 0.0f;

  const float* Kbase = Km + (size_t)(b * N) * F + h * D;
  const float* Vbase = Vm + (size_t)(b * N) * F + h * D;

  // per-thread slice of the cooperative 32-key chunk copy: 8 floats per array
  const int crr = threadIdx.x >> 3;       // 0..31 : key row
  const int ccc = (threadIdx.x & 7) * 8;  // 0..56 : dim offset

  auto issue_chunk = [&](int kc, int buf) {
    const float* kg = Kbase + (size_t)(kc + crr) * F + ccc;
    const float* vg = Vbase + (size_t)(kc + crr) * F + ccc;
    float* ks = &Kld[buf][crr * KSTR + ccc];
    float* vs = &Vld[buf][crr * KSTR + ccc];
    async_cp16(ks, kg);
    async_cp16(ks + 4, kg + 4);
    async_cp16(vs, vg);
    async_cp16(vs + 4, vg + 4);
  };

  issue_chunk(0, 0);

  int it = 0;
  for (int kc = 0; kc < N; kc += CH, ++it) {
    const int cur = it & 1;
    const bool have_next = (kc + CH) < N;
    if (have_next) {
      issue_chunk(kc + CH, cur ^ 1);  // overwrites buffer consumed at it-1
      ASYNC_WAIT_PREV();              // wait only for current chunk (in-order)
    } else {
      ASYNC_WAIT_ALL();
    }
    __syncthreads();  // current chunk visible to all waves

#pragma unroll 1
    for (int ksub = 0; ksub < 2; ++ksub) {
      // S = (Q*scale) @ K_tile^T : M=16 q-rows, N=16 keys, K=64
      const float* Kt = &Kld[cur][(ksub * 16 + ln) * KSTR + 4 * hl];
      v4f kf[8];
#pragma unroll
      for (int kb = 0; kb < 8; ++kb) kf[kb] = *(const v4f*)(Kt + kb * 8);
      __builtin_amdgcn_sched_barrier(0);  // all 8 ds_load_b128 before WMMAs
      v8f s0 = vzero8(), s1 = vzero8();
#pragma unroll
      for (int kb = 0; kb < 8; ++kb) {
        s0 = wmma4(lo2(qa4[kb]), lo2(kf[kb]), s0);
        s1 = wmma4(hi2(qa4[kb]), hi2(kf[kb]), s1);
      }
      v8f s = s0 + s1;

      // C/D layout -> LDS row-major P[m][n]
      float* Pw = Pld[wid];
#pragma unroll
      for (int j = 0; j < 8; ++j) Pw[(j + 8 * hl) * PSTR + ln] = s[j];
      WAVE_LDS_FENCE();

      // online softmax: lane r (0..15) owns q-row r
      if (lane < 16) {
        float* pr = Pw + lane * PSTR;
        float tm = pr[0];
#pragma unroll
        for (int n = 1; n < 16; ++n) tm = fmaxf(tm, pr[n]);
        const float mnew = fmaxf(mi, tm);
        const float corr = __expf(mi - mnew);
        float ssum = 0.0f;
#pragma unroll
        for (int n = 0; n < 16; ++n) {
          float p = __expf(pr[n] - mnew);
          pr[n] = p;
          ssum += p;
        }
        li = li * corr + ssum;
        mi = mnew;
        Stat[wid][lane] = corr;
      }
      WAVE_LDS_FENCE();

      // rescale running O by per-row correction
      float crf[8];
#pragma unroll
      for (int j = 0; j < 8; ++j) crf[j] = Stat[wid][j + 8 * hl];
#pragma unroll
      for (int vb = 0; vb < 4; ++vb)
#pragma unroll
        for (int j = 0; j < 8; ++j) oacc[vb][j] *= crf[j];

      // O += P(16x16) @ V_tile(16x64): A = P from LDS, B = V from LDS
      const float* Pwm = Pw + ln * PSTR + 4 * hl;
      v4f af[2];
      af[0] = *(const v4f*)(Pwm + 0);
      af[1] = *(const v4f*)(Pwm + 8);
#pragma unroll
      for (int vb = 0; vb < 4; ++vb) {
#pragma unroll
        for (int kb = 0; kb < 2; ++kb) {
          const int r0 = ksub * 16 + kb * 8 + 4 * hl;
          v2f b0, b1;
          b0.x = Vld[cur][(r0 + 0) * KSTR + vb * 16 + ln];
          b0.y = Vld[cur][(r0 + 1) * KSTR + vb * 16 + ln];
          b1.x = Vld[cur][(r0 + 2) * KSTR + vb * 16 + ln];
          b1.y = Vld[cur][(r0 + 3) * KSTR + vb * 16 + ln];
          oacc[vb] = wmma4(lo2(af[kb]), b0, oacc[vb]);
          oacc[vb] = wmma4(hi2(af[kb]), b1, oacc[vb]);
        }
      }
    }
    __syncthreads();  // all waves done with buf[cur] before it is re-filled
  }

  // finalize: divide by row sums, write cat[b*N+row][h*64 + d]
  if (lane < 16) Stat[wid][lane] = 1.0f / li;
  WAVE_LDS_FENCE();
  float inv[8];
#pragma unroll
  for (int j = 0; j < 8; ++j) inv[j] = Stat[wid][j + 8 * hl];
#pragma unroll
  for (int vb = 0; vb < 4; ++vb)
#pragma unroll
    for (int j = 0; j < 8; ++j) {
      const int m = j + 8 * hl;
      Cat[(size_t)(b * N + qbase + m) * F + h * D + vb * 16 + ln] =
          oacc[vb][j] * inv[j];
    }
}

// ---------------------------------------------------------------------------
// Stage 3: Y = cat @ Wp^T, fused BatchNorm (inference) + LeakyReLU
// cat:[16384][512], Wp:[256][512], Y:[16384][256].  64x16 strip per wave.
// ---------------------------------------------------------------------------
__global__ __launch_bounds__(256) void out_proj_bn(const float* __restrict__ Cat,
                                                   const float* __restrict__ Wp,
                                                   const float* __restrict__ gamma,
                                                   const float* __restrict__ beta,
                                                   const float* __restrict__ mean,
                                                   const float* __restrict__ var,
                                                   float* __restrict__ Out) {
  constexpr int NC = 256, K = 512;
  const int lane = threadIdx.x & 31;
  const int wid  = threadIdx.x >> 5;
  const int ln   = lane & 15;
  const int hl   = lane >> 4;

  const int ntiles = NC >> 4;  // 16
  const int gw = blockIdx.x * 8 + wid;
  const int mt = gw / ntiles;  // 64-row strip
  const int nt = gw % ntiles;

  const float* ar0 = Cat + (size_t)(mt * 64 + 0 * 16 + ln) * K;
  const float* ar1 = Cat + (size_t)(mt * 64 + 1 * 16 + ln) * K;
  const float* ar2 = Cat + (size_t)(mt * 64 + 2 * 16 + ln) * K;
  const float* ar3 = Cat + (size_t)(mt * 64 + 3 * 16 + ln) * K;
  const float* brow = Wp + (size_t)(nt * 16 + ln) * K;

  v8f acc[4] = {vzero8(), vzero8(), vzero8(), vzero8()};
#pragma unroll 2
  for (int kb = 0; kb < (K >> 3); kb += 2) {
    v4f bb[2];
    v4f aa[2][4];
#pragma unroll
    for (int s = 0; s < 2; ++s) {
      const int ko = (kb + s) * 8 + 4 * hl;
      bb[s]    = *(const v4f*)(brow + ko);
      aa[s][0] = *(const v4f*)(ar0 + ko);
      aa[s][1] = *(const v4f*)(ar1 + ko);
      aa[s][2] = *(const v4f*)(ar2 + ko);
      aa[s][3] = *(const v4f*)(ar3 + ko);
    }
    __builtin_amdgcn_sched_barrier(0);
#pragma unroll
    for (int s = 0; s < 2; ++s) {
#pragma unroll
      for (int i = 0; i < 4; ++i)
        acc[i] = wmma4(lo2(aa[s][i]), lo2(bb[s]), acc[i]);
#pragma unroll
      for (int i = 0; i < 4; ++i)
        acc[i] = wmma4(hi2(aa[s][i]), hi2(bb[s]), acc[i]);
    }
  }

  const int c = nt * 16 + ln;  // output channel, constant per lane
  const float sc = gamma[c] * rsqrtf(var[c] + 1e-5f);
  const float mu = mean[c];
  const float bt = beta[c];

#pragma unroll
  for (int i = 0; i < 4; ++i) {
    float* orow = Out + (size_t)(mt * 64 + i * 16 + hl * 8) * NC + c;
#pragma unroll
    for (int j = 0; j < 8; ++j) {
      float y = (acc[i][j] - mu) * sc + bt;
      y = (y >= 0.0f) ? y : 0.01f * y;
      orow[(size_t)j * NC] = y;
    }
  }
}

// ---------------------------------------------------------------------------
extern "C" void kernel_launch(void* const* d_in, const int* in_sizes, int n_in,
                              void* d_out, int out_size, void* d_ws,
                              size_t ws_size, hipStream_t stream) {
  (void)in_sizes; (void)n_in; (void)out_size; (void)ws_size;
  const float* x     = (const float*)d_in[0];
  const float* Wq    = (const float*)d_in[1];
  const float* Wk    = (const float*)d_in[2];
  const float* Wv    = (const float*)d_in[3];
  const float* Wp    = (const float*)d_in[4];
  const float* gamma = (const float*)d_in[5];
  const float* beta  = (const float*)d_in[6];
  const float* mean  = (const float*)d_in[7];
  const float* var   = (const float*)d_in[8];
  float* out = (float*)d_out;

  float* ws = (float*)d_ws;
  const size_t SZ = (size_t)16384 * 512;  // q/k/v/cat each
  float* q   = ws;
  float* k   = ws + SZ;
  float* v   = ws + 2 * SZ;
  float* cat = ws + 3 * SZ;

  dim3 blk(256);
  // QKV projections: (16384/64)*(512/16) = 8192 waves / 8 = 1024 blocks each
  gemm_xt<<<1024, blk, 0, stream>>>(x, Wq, q, 16384, 512, 256);
  gemm_xt<<<1024, blk, 0, stream>>>(x, Wk, k, 16384, 512, 256);
  gemm_xt<<<1024, blk, 0, stream>>>(x, Wv, v, 16384, 512, 256);
  // attention: 64 (b,h) * 16 q-blocks = 1024 workgroups
  flash_attn<<<1024, blk, 0, stream>>>(q, k, v, cat);
  // output projection + BN + LeakyReLU: (16384/64)*(256/16)/8 = 512 blocks
  out_proj_bn<<<512, blk, 0, stream>>>(cat, Wp, gamma, beta, mean, var, out);
}